// HotProposalLayer_30339648978947
// MI455X (gfx1250) — compile-verified
//
#include <hip/hip_runtime.h>
#include <stdint.h>

#define B_ 8
#define N_ 50000
#define C_ 256
#define H_ 160
#define W_ 160
#define HW_ (H_ * W_)
#define S_ 161   // SAT side (exclusive pad)

typedef float v2f __attribute__((ext_vector_type(2)));
typedef float v8f __attribute__((ext_vector_type(8)));

// ---------------- K1: per-(b,c) spatial mean -------------------------------
__global__ void k_mean(const float* __restrict__ f, float* __restrict__ mean) {
    int bc = blockIdx.x;                       // b*C + c
    const float* p = f + (size_t)bc * HW_;
    float s = 0.f;
    for (int i = threadIdx.x; i < HW_; i += 256) s += p[i];
    __shared__ float wsum[8];
    int lane = threadIdx.x & 31, w = threadIdx.x >> 5;
    for (int o = 16; o > 0; o >>= 1) s += __shfl_down(s, o, 32);
    if (lane == 0) wsum[w] = s;
    __syncthreads();
    if (w == 0) {
        s = (lane < 8) ? wsum[lane] : 0.f;
        for (int o = 4; o > 0; o >>= 1) s += __shfl_down(s, o, 32);
        if (lane == 0) mean[bc] = s / (float)HW_;
    }
}

// ---------------- K2: fmap = sum_c |f - mean_c| via WMMA ones-matmul -------
// One wave handles 16 pixels; each V_WMMA_F32_16X16X4_F32 folds 4 channels:
//   D = ones(16x4) x |diff|(4x16) + C  ->  every row of D = per-pixel sum.
__global__ void k_fmap_wmma(const float* __restrict__ f,
                            const float* __restrict__ mean,
                            float* __restrict__ fmap) {
    int b    = blockIdx.x / (HW_ / 128);
    int tile = blockIdx.x % (HW_ / 128);
    __shared__ float sm[C_];
    sm[threadIdx.x] = mean[b * C_ + threadIdx.x];
    __syncthreads();

    int wave = threadIdx.x >> 5;
    int lane = threadIdx.x & 31;
    int half = lane >> 4;                       // which K row this lane feeds
    int pos  = tile * 128 + wave * 16 + (lane & 15);
    const float* fb = f + (size_t)b * C_ * HW_;

    v2f aone; aone[0] = 1.f; aone[1] = 1.f;     // all-ones A matrix
    v8f acc = {};
    for (int c = 0; c < C_; c += 4) {
        float d0 = fabsf(fb[(size_t)(c +     half) * HW_ + pos] - sm[c +     half]);
        float d1 = fabsf(fb[(size_t)(c + 2 + half) * HW_ + pos] - sm[c + 2 + half]);
        v2f bv; bv[0] = d0; bv[1] = d1;
        acc = __builtin_amdgcn_wmma_f32_16x16x4_f32(
            false, aone, false, bv, (short)0, acc, false, false);
    }
    if (half == 0) fmap[(size_t)b * HW_ + pos] = acc[0];  // D row 0, N = lane
}

// ---------------- K3a: row-wise inclusive scan (wave32 shuffle scan) -------
__global__ void k_rowscan(const float* __restrict__ fmap, float* __restrict__ sat) {
    int r = blockIdx.x * 8 + (threadIdx.x >> 5);  // 1280 rows total
    int lane = threadIdx.x & 31;
    int b = r / H_, y = r % H_;
    const float* row = fmap + (size_t)(b * H_ + y) * W_;
    float* srow = sat + (size_t)(b * S_ + (y + 1)) * S_;
    if (lane == 0) srow[0] = 0.f;                 // exclusive pad column
    float carry = 0.f;
    for (int chunk = 0; chunk < W_; chunk += 32) {
        float v = row[chunk + lane];
        for (int o = 1; o < 32; o <<= 1) {
            float t = __shfl_up(v, o, 32);
            if (lane >= o) v += t;
        }
        v += carry;
        srow[1 + chunk + lane] = v;
        carry = __shfl(v, 31, 32);
    }
}

// ---------------- K3b: column-wise scan, register carry, coalesced ---------
__global__ void k_colscan(float* __restrict__ sat) {
    int b = blockIdx.x;
    int x = threadIdx.x;
    if (x > 160) return;
    float* s = sat + (size_t)b * S_ * S_;
    s[x] = 0.f;                                   // exclusive pad row
    float c = 0.f;
    for (int y = 1; y <= 160; y += 4) {           // unroll to pipeline loads
        float v0 = s[(y + 0) * S_ + x];
        float v1 = s[(y + 1) * S_ + x];
        float v2 = s[(y + 2) * S_ + x];
        float v3 = s[(y + 3) * S_ + x];
        c += v0; s[(y + 0) * S_ + x] = c;
        c += v1; s[(y + 1) * S_ + x] = c;
        c += v2; s[(y + 2) * S_ + x] = c;
        c += v3; s[(y + 3) * S_ + x] = c;
    }
}

// ---------------- K4: per-anchor heat -> 64-bit sortable key ---------------
__global__ void k_heat(const float* __restrict__ anchors,
                       const float* __restrict__ sat,
                       unsigned long long* __restrict__ keys) {
    int i = blockIdx.x * blockDim.x + threadIdx.x;
    if (i >= B_ * N_) return;
    int b = i / N_;
    int n = i - b * N_;
    float4 a = ((const float4*)anchors)[i];
    int y1 = (int)rintf(a.x * (float)H_);         // RNE matches jnp.round
    int x1 = (int)rintf(a.y * (float)W_);
    int y2 = (int)rintf(a.z * (float)H_);
    int x2 = (int)rintf(a.w * (float)W_);
    const float* s = sat + (size_t)b * S_ * S_;
    float box = s[y2 * S_ + x2] - s[y1 * S_ + x2]
              - s[y2 * S_ + x1] + s[y1 * S_ + x1];
    float area = (float)((y2 - y1) * (x2 - x1));
    float h = box / area;
    unsigned u = __float_as_uint(h);
    u = (u & 0x80000000u) ? ~u : (u | 0x80000000u);   // order-preserving map
    keys[i] = ((unsigned long long)u << 32) | (unsigned)(~(unsigned)n);
}

// ---------------- K5: per-batch 64-bit radix select (kth largest key) ------
__global__ void k_select(const unsigned long long* __restrict__ keys,
                         unsigned long long* __restrict__ thresh,
                         unsigned* __restrict__ cnt, int K) {
    int b = blockIdx.x;
    const unsigned long long* kb = keys + (size_t)b * N_;
    __shared__ unsigned hist[256];
    __shared__ unsigned long long prefix;
    __shared__ int kk;
    if (threadIdx.x == 0) { prefix = 0ull; kk = K; }
    __syncthreads();
    for (int pass = 7; pass >= 0; --pass) {
        hist[threadIdx.x] = 0u;                    // blockDim == 256
        __syncthreads();
        unsigned long long pf = prefix;
        int shift = pass * 8;
        for (int i = threadIdx.x; i < N_; i += 256) {
            unsigned long long key = kb[i];
            bool match = (pass == 7) ||
                         ((key >> (shift + 8)) == (pf >> (shift + 8)));
            if (match) atomicAdd(&hist[(unsigned)(key >> shift) & 0xFFu], 1u);
        }
        __syncthreads();
        if (threadIdx.x == 0) {
            unsigned cum = 0; int target = kk;
            for (int d = 255; d >= 0; --d) {
                unsigned c = hist[d];
                if (cum + c >= (unsigned)target) {
                    prefix |= ((unsigned long long)(unsigned)d) << shift;
                    kk = target - (int)cum;
                    break;
                }
                cum += c;
            }
        }
        __syncthreads();
    }
    if (threadIdx.x == 0) { thresh[b] = prefix; cnt[b] = 0u; }
}

// ---------------- K6: compact the K winners (unordered) --------------------
__global__ void k_compact(const unsigned long long* __restrict__ keys,
                          const unsigned long long* __restrict__ thresh,
                          unsigned* __restrict__ cnt,
                          unsigned long long* __restrict__ selkey,
                          int* __restrict__ selidx, int K) {
    int i = blockIdx.x * blockDim.x + threadIdx.x;
    if (i >= B_ * N_) return;
    int b = i / N_;
    unsigned long long key = keys[i];
    if (key >= thresh[b]) {                        // exactly K per batch (unique keys)
        unsigned pos = atomicAdd(&cnt[b], 1u);
        if (pos < (unsigned)K) {
            selkey[(size_t)b * K + pos] = key;
            selidx[(size_t)b * K + pos] = i - b * N_;
        }
    }
}

// ---------------- K7: O(K^2) rank in LDS, emit anchors in top-k order ------
__global__ void k_emit(const unsigned long long* __restrict__ selkey,
                       const int* __restrict__ selidx,
                       const float* __restrict__ anchors,
                       float* __restrict__ out, int K) {
    int b = blockIdx.x;
    __shared__ unsigned long long sk[1024];        // K <= 1024 (K == 1000)
    for (int j = threadIdx.x; j < K; j += blockDim.x)
        sk[j] = selkey[(size_t)b * K + j];
    __syncthreads();
    for (int j = threadIdx.x; j < K; j += blockDim.x) {
        unsigned long long mykey = sk[j];
        int rank = 0;
        for (int t = 0; t < K; ++t) rank += (sk[t] > mykey) ? 1 : 0;
        int n = selidx[(size_t)b * K + j];
        float4 a = ((const float4*)anchors)[(size_t)b * N_ + n];
        ((float4*)out)[(size_t)b * K + rank] = a;
    }
}

extern "C" void kernel_launch(void* const* d_in, const int* in_sizes, int n_in,
                              void* d_out, int out_size, void* d_ws, size_t ws_size,
                              hipStream_t stream) {
    const float* anchors = (const float*)d_in[0];
    const float* fmapsrc = (const float*)d_in[1];
    (void)in_sizes; (void)n_in; (void)ws_size;
    int K = out_size / (B_ * 4);                   // == counts (1000)

    char* ws = (char*)d_ws;
    const size_t off_mean   = 0;                                   //   8,192 B
    const size_t off_fmap   = off_mean   + (size_t)B_ * C_ * 4;    // 819,200 B
    const size_t off_sat    = off_fmap   + (size_t)B_ * HW_ * 4;   // 829,472 B
    const size_t off_keys   = off_sat    + (size_t)B_ * S_ * S_ * 4; // 3.2 MB
    const size_t off_thresh = off_keys   + (size_t)B_ * N_ * 8;
    const size_t off_cnt    = off_thresh + (size_t)B_ * 8;
    const size_t off_selkey = off_cnt    + 64;                     // 8B-aligned
    const size_t off_selidx = off_selkey + (size_t)B_ * K * 8;

    float*              mean   = (float*)(ws + off_mean);
    float*              fmap   = (float*)(ws + off_fmap);
    float*              sat    = (float*)(ws + off_sat);
    unsigned long long* keys   = (unsigned long long*)(ws + off_keys);
    unsigned long long* thresh = (unsigned long long*)(ws + off_thresh);
    unsigned*           cnt    = (unsigned*)(ws + off_cnt);
    unsigned long long* selkey = (unsigned long long*)(ws + off_selkey);
    int*                selidx = (int*)(ws + off_selidx);

    k_mean     <<<B_ * C_,          256, 0, stream>>>(fmapsrc, mean);
    k_fmap_wmma<<<B_ * (HW_ / 128), 256, 0, stream>>>(fmapsrc, mean, fmap);
    k_rowscan  <<<B_ * H_ / 8,      256, 0, stream>>>(fmap, sat);
    k_colscan  <<<B_,               192, 0, stream>>>(sat);
    int tot = B_ * N_;
    k_heat     <<<(tot + 255) / 256, 256, 0, stream>>>(anchors, sat, keys);
    k_select   <<<B_,               256, 0, stream>>>(keys, thresh, cnt, K);
    k_compact  <<<(tot + 255) / 256, 256, 0, stream>>>(keys, thresh, cnt, selkey, selidx, K);
    k_emit     <<<B_,               256, 0, stream>>>(selkey, selidx, anchors, (float*)d_out, K);
}